// Keypoint_88562225643950
// MI455X (gfx1250) — compile-verified
//
#include <hip/hip_runtime.h>
#include <stdint.h>

#define W 512
#define PATCH 32
#define NPP 256      // patches per image (16x16)
#define NCH 128
#define NB 8

__device__ __forceinline__ float waveSum(float v) {
#pragma unroll
    for (int o = 16; o > 0; o >>= 1) v += __shfl_xor(v, o, 32);
    return v;
}
__device__ __forceinline__ float waveMax(float v) {
#pragma unroll
    for (int o = 16; o > 0; o >>= 1) v = fmaxf(v, __shfl_xor(v, o, 32));
    return v;
}

// CDNA5 async global->LDS copy, 16B per lane (tracked by ASYNCcnt).
__device__ __forceinline__ void async_ld_b128(uint32_t lds_off, uint64_t gaddr) {
    asm volatile("global_load_async_to_lds_b128 %0, %1, off"
                 :: "v"(lds_off), "v"(gaddr) : "memory");
}

__global__ __launch_bounds__(256) void keypoint_kernel(
    const float* __restrict__ det, const float* __restrict__ wsc,
    const float* __restrict__ desc, float* __restrict__ out)
{
    __shared__ float att[1024];
    __shared__ __align__(16) float stage[8][3][256];   // per-wave triple buffer, 1KB each
    __shared__ float red[8];
    __shared__ float4 red4[8];

    const int blk = blockIdx.x;
    const int b = blk >> 8;
    const int l = blk & 255;
    const int py = l >> 4, px = l & 15;
    const int t = threadIdx.x;
    const int lane = t & 31, wid = t >> 5;

    // -------- descriptor async-staging setup (independent of softmax) --------
    const int cbase = wid * 16;                  // 16 channels per wave
    const int q_row = lane >> 3, q_col = lane & 7;
    const uint64_t descbase = (uint64_t)(uintptr_t)desc;
    uint32_t stage_off[3];
    stage_off[0] = (uint32_t)(uintptr_t)&stage[wid][0][0];
    stage_off[1] = (uint32_t)(uintptr_t)&stage[wid][1][0];
    stage_off[2] = (uint32_t)(uintptr_t)&stage[wid][2][0];

    // one iteration = one 1KB chunk (8 rows x 32 cols) of one channel's patch
    auto issue = [&](int it, int sel) {
        int ch = cbase + (it >> 2);
        int tch = it & 3;
#pragma unroll
        for (int q = 0; q < 2; ++q) {
            int r = tch * 8 + q * 4 + q_row;     // row within 32x32 patch
            uint64_t eoff = (((uint64_t)(b * NCH + ch) * W) + (uint64_t)(py * PATCH + r)) * W
                            + (uint64_t)(px * PATCH + q_col * 4);
            uint32_t loff = stage_off[sel] + (uint32_t)((q * 4 + q_row) * 128 + q_col * 16);
            async_ld_b128(loff, descbase + eoff * 4ull);
        }
    };

    // kick off the first two chunks now; async engine streams while we do softmax
    issue(0, 0);
    issue(1, 1);

    // ---------------- Phase 1: softmax over detector patch ----------------
    // thread t handles 4 contiguous pixels k = 4t..4t+3 (one b128 per input)
    const size_t imgbase = (size_t)b * (W * W);
    const int ki = t >> 3;               // patch row 0..31
    const int kjb = (t & 7) * 4;         // patch col base
    const int gy = py * PATCH + ki;
    const float gxb = (float)(px * PATCH + kjb);
    const size_t rowoff = imgbase + (size_t)gy * W + (size_t)(px * PATCH + kjb);
    float4 dv = *reinterpret_cast<const float4*>(det + rowoff);
    float4 wv = *reinterpret_cast<const float4*>(wsc + rowoff);
    float d[4] = {dv.x, dv.y, dv.z, dv.w};
    float w[4] = {wv.x, wv.y, wv.z, wv.w};

    float m = fmaxf(fmaxf(d[0], d[1]), fmaxf(d[2], d[3]));
    m = waveMax(m);
    if (lane == 0) red[wid] = m;
    __syncthreads();
    if (wid == 0) {
        float mm = (lane < 8) ? red[lane] : -3.4e38f;
        mm = waveMax(mm);
        if (lane == 0) red[0] = mm;
    }
    __syncthreads();
    m = red[0];

    float e[4];
    float se = 0.f, su = 0.f, sv = 0.f, sw = 0.f;
#pragma unroll
    for (int j = 0; j < 4; ++j) {
        e[j] = __expf(d[j] - m);
        se += e[j];
        su += e[j] * (gxb + (float)j);   // expected_u: column coordinate
        sw += e[j] * w[j];
    }
    sv = se * (float)gy;                 // expected_v: row coordinate (const per thread)
    se = waveSum(se); su = waveSum(su); sv = waveSum(sv); sw = waveSum(sw);
    if (lane == 0) red4[wid] = make_float4(se, su, sv, sw);
    __syncthreads();
    if (wid == 0) {
        float4 z = (lane < 8) ? red4[lane] : make_float4(0.f, 0.f, 0.f, 0.f);
        z.x = waveSum(z.x); z.y = waveSum(z.y);
        z.z = waveSum(z.z); z.w = waveSum(z.w);
        if (lane == 0) red4[0] = z;
    }
    __syncthreads();
    float4 s = red4[0];
    float inv = 1.0f / s.x;
    *reinterpret_cast<float4*>(att + 4 * t) =
        make_float4(e[0] * inv, e[1] * inv, e[2] * inv, e[3] * inv);
    if (t == 0) {
        out[(size_t)(b * NPP + l) * 2 + 0] = s.y * inv;           // coords u
        out[(size_t)(b * NPP + l) * 2 + 1] = s.z * inv;           // coords v
        out[(size_t)(NB * NPP * 2) + b * NPP + l] = s.w * inv;    // score
    }
    __syncthreads();

    // ---------------- Phase 2: descriptor reduction (depth-2 async pipeline) ----------------
    // Each lane keeps its att column in registers: attc[r] = att[r*32 + lane]
    float attc[32];
#pragma unroll
    for (int r = 0; r < 32; ++r) attc[r] = att[r * 32 + lane];

    float* outd = out + (NB * NPP * 2 + NB * NPP);
    float acc = 0.0f;
    int sel = 0, selN = 2;               // buffer of chunk it, buffer of chunk it+2
    for (int it = 0; it < 64; ++it) {
        if (it + 2 < 64) {
            issue(it + 2, selN);
            // chunks it..it+2 in flight (6 ops); <=4 outstanding => chunk it landed
            asm volatile("s_wait_asynccnt 0x4" ::: "memory");
        } else if (it + 1 < 64) {
            asm volatile("s_wait_asynccnt 0x2" ::: "memory");
        } else {
            asm volatile("s_wait_asynccnt 0x0" ::: "memory");
        }
        const float* bp = &stage[wid][sel][0];
        int tch = it & 3;
#pragma unroll
        for (int i = 0; i < 8; ++i)
            acc = fmaf(bp[i * 32 + lane], attc[tch * 8 + i], acc);
        if (tch == 3) {
            float r = waveSum(acc);
            if (lane == 0) {
                int ch = cbase + (it >> 2);
                outd[(size_t)(b * NCH + ch) * NPP + l] = r;
            }
            acc = 0.0f;
        }
        sel  = (sel  == 2) ? 0 : sel  + 1;
        selN = (selN == 2) ? 0 : selN + 1;
    }
}

extern "C" void kernel_launch(void* const* d_in, const int* in_sizes, int n_in,
                              void* d_out, int out_size, void* d_ws, size_t ws_size,
                              hipStream_t stream) {
    const float* det  = (const float*)d_in[0];   // (8,1,512,512) f32
    const float* wsc  = (const float*)d_in[1];   // (8,1,512,512) f32
    const float* desc = (const float*)d_in[2];   // (8,128,512,512) f32
    float* out = (float*)d_out;                  // coords | scores | desc, flat f32

    dim3 grid(NB * NPP);   // 2048 blocks: one per (batch, patch)
    dim3 block(256);       // 8 wave32
    hipLaunchKernelGGL(keypoint_kernel, grid, block, 0, stream, det, wsc, desc, out);
}